// ScaledDotProductAttention_57183194579668
// MI455X (gfx1250) — compile-verified
//
#include <hip/hip_runtime.h>
#include <stdint.h>

typedef __attribute__((ext_vector_type(2))) float v2f;
typedef __attribute__((ext_vector_type(8))) float v8f;

#define S_LEN     2048
#define D_HEAD    64
#define QT        16                 // query rows per workgroup
#define KT        64                 // K/V rows staged per LDS chunk
#define NTHREADS  128                // 4 waves (wave32)
#define SC_STRIDE (S_LEN + 4)        // 2052 floats: row stride 8208B = 16B-aligned, bank offset 4/row
#define KV_STRIDE (D_HEAD + 4)       // 68 floats:  row stride 272B  = 16B-aligned, bank-skewed

__global__ __launch_bounds__(NTHREADS)
void attn_fwd_f32wmma_kernel(const float* __restrict__ q,
                             const float* __restrict__ kmat,
                             const float* __restrict__ vmat,
                             const unsigned char* __restrict__ mask,
                             float* __restrict__ out_wv,
                             float* __restrict__ out_probs)
{
    __shared__ float sc[QT][SC_STRIDE];      // 131,328 B: full 16x2048 score/prob block
    __shared__ float qtile[QT][KV_STRIDE];   //   4,352 B (pre-scaled by 1/sqrt(d))
    __shared__ float kv[KT][KV_STRIDE];      //  17,408 B: K or V chunk
    __shared__ float red[NTHREADS];
    __shared__ float rowmax[QT];
    __shared__ float rowinv[QT];

    const int tid  = threadIdx.x;
    const int wave = tid >> 5;
    const int lane = tid & 31;
    const int m    = lane & 15;       // M (A rows) / N (B,C,D cols) per WMMA 16x16x4 layout
    const int h    = lane >> 4;       // lane half: K = j + 2h for A/B, row = i + 8h for C/D
    const int wcol = wave << 4;       // 16-column slice owned by this wave

    const int bh = blockIdx.x >> 7;                 // 128 query tiles per (b,h)
    const int qb = (blockIdx.x & 127) << 4;

    const size_t qkv_base = (size_t)bh * S_LEN * D_HEAD;
    const float  scale    = 0.125f;                 // 1/sqrt(64), folded into Q tile

    // ---------------- load Q tile (16x64, coalesced float4), pre-scaled ----------------
    {
        const float4* src = (const float4*)(q + qkv_base + (size_t)qb * D_HEAD);
        #pragma unroll
        for (int it = 0; it < (QT * D_HEAD) / 4 / NTHREADS; ++it) {
            int e = tid + it * NTHREADS;
            float4 val = src[e];
            val.x *= scale; val.y *= scale; val.z *= scale; val.w *= scale;
            int g = e * 4, r = g >> 6, c = g & 63;
            *(float4*)&qtile[r][c] = val;
        }
    }

    // ---------------- phase 1: scores = (Q/sqrt(d)) K^T via v_wmma_f32_16x16x4_f32 ----------------
    for (int kb0 = 0; kb0 < S_LEN; kb0 += KT) {
        __syncthreads();                              // prior chunk's WMMA reads done
        {
            const float4* src = (const float4*)(kmat + qkv_base + (size_t)kb0 * D_HEAD);
            #pragma unroll
            for (int it = 0; it < (KT * D_HEAD) / 4 / NTHREADS; ++it) {
                int e = tid + it * NTHREADS;
                float4 val = src[e];
                int g = e * 4, r = g >> 6, c = g & 63;
                *(float4*)&kv[r][c] = val;
            }
            if (kb0 + KT < S_LEN)                     // pull next 16KB chunk toward L2
                __builtin_prefetch(kmat + qkv_base + (size_t)(kb0 + KT) * D_HEAD + tid * 32, 0, 1);
        }
        __syncthreads();

        // two independent accumulator chains (even/odd K-steps) -> 2x WMMA ILP
        v8f acc0 = {}, acc1 = {};
        #pragma unroll
        for (int kk = 0; kk < D_HEAD / 4; kk += 2) {
            v2f a0 = *(const v2f*)&qtile[m][4 * kk + 2 * h];
            v2f b0 = *(const v2f*)&kv[wcol + m][4 * kk + 2 * h];
            v2f a1 = *(const v2f*)&qtile[m][4 * (kk + 1) + 2 * h];
            v2f b1 = *(const v2f*)&kv[wcol + m][4 * (kk + 1) + 2 * h];
            acc0 = __builtin_amdgcn_wmma_f32_16x16x4_f32(false, a0, false, b0,
                                                         (short)0, acc0, false, false);
            acc1 = __builtin_amdgcn_wmma_f32_16x16x4_f32(false, a1, false, b1,
                                                         (short)0, acc1, false, false);
        }
        #pragma unroll
        for (int i = 0; i < 8; ++i)                   // C/D layout: vgpr i, half h -> row i+8h
            sc[i + 8 * h][kb0 + wcol + m] = acc0[i] + acc1[i];
    }
    __syncthreads();

    // ---------------- mask pass (coalesced 4-byte packed bool loads) ----------------
    {
        const uint32_t* m4 = (const uint32_t*)(mask + ((size_t)bh * S_LEN + qb) * S_LEN);
        for (int e = tid; e < (QT * S_LEN) / 4; e += NTHREADS) {
            uint32_t mv = m4[e];
            if (mv) {
                int g = e * 4, r = g >> 11, c = g & (S_LEN - 1);
                if (mv & 0x000000FFu) sc[r][c + 0] = -INFINITY;
                if (mv & 0x0000FF00u) sc[r][c + 1] = -INFINITY;
                if (mv & 0x00FF0000u) sc[r][c + 2] = -INFINITY;
                if (mv & 0xFF000000u) sc[r][c + 3] = -INFINITY;
            }
        }
    }
    __syncthreads();

    // ---------------- softmax stats: 8 threads per row ----------------
    {
        const int r = tid >> 3, j = tid & 7;
        float lm = -INFINITY;
        for (int c = j; c < S_LEN; c += 8) lm = fmaxf(lm, sc[r][c]);
        red[tid] = lm;
        __syncthreads();
        if (j == 0) {
            float mx = red[tid];
            #pragma unroll
            for (int t = 1; t < 8; ++t) mx = fmaxf(mx, red[tid + t]);
            rowmax[r] = mx;
        }
        __syncthreads();
        const float rm = rowmax[r];
        float ls = 0.f;
        for (int c = j; c < S_LEN; c += 8) {
            float e = __expf(sc[r][c] - rm);
            sc[r][c] = e;
            ls += e;
        }
        red[tid] = ls;
        __syncthreads();
        if (j == 0) {
            float s = 0.f;
            #pragma unroll
            for (int t = 0; t < 8; ++t) s += red[tid + t];
            rowinv[r] = 1.0f / s;
        }
        __syncthreads();
    }

    // ---------------- normalize + store probs: dominant HBM traffic, float4 / b128 ----------------
    {
        float4* probs4 = (float4*)(out_probs + ((size_t)bh * S_LEN + qb) * S_LEN);
        for (int e4 = tid; e4 < (QT * S_LEN) / 4; e4 += NTHREADS) {
            int g = e4 * 4, r = g >> 11, c = g & (S_LEN - 1);
            float inv = rowinv[r];
            float4 pv = *(const float4*)&sc[r][c];    // 16B-aligned (stride 8208B)
            pv.x *= inv; pv.y *= inv; pv.z *= inv; pv.w *= inv;
            *(float4*)&sc[r][c] = pv;
            probs4[e4] = pv;                          // 2KB contiguous per iteration
        }
    }

    // ---------------- phase 3: O = P V via v_wmma_f32_16x16x4_f32, dual chains ----------------
    v8f oacc0 = {}, oacc1 = {};
    for (int vb0 = 0; vb0 < S_LEN; vb0 += KT) {
        __syncthreads();
        {
            const float4* src = (const float4*)(vmat + qkv_base + (size_t)vb0 * D_HEAD);
            #pragma unroll
            for (int it = 0; it < (KT * D_HEAD) / 4 / NTHREADS; ++it) {
                int e = tid + it * NTHREADS;
                float4 val = src[e];
                int g = e * 4, r = g >> 6, c = g & 63;
                *(float4*)&kv[r][c] = val;
            }
            if (vb0 + KT < S_LEN)
                __builtin_prefetch(vmat + qkv_base + (size_t)(vb0 + KT) * D_HEAD + tid * 32, 0, 1);
        }
        __syncthreads();
        #pragma unroll
        for (int kk = 0; kk < KT / 4; kk += 2) {
            int kc0 = kk * 4, kc1 = (kk + 1) * 4;
            v2f a0 = *(const v2f*)&sc[m][vb0 + kc0 + 2 * h];
            v2f b0; b0.x = kv[kc0 + 2 * h][wcol + m];
                    b0.y = kv[kc0 + 1 + 2 * h][wcol + m];
            v2f a1 = *(const v2f*)&sc[m][vb0 + kc1 + 2 * h];
            v2f b1; b1.x = kv[kc1 + 2 * h][wcol + m];
                    b1.y = kv[kc1 + 1 + 2 * h][wcol + m];
            oacc0 = __builtin_amdgcn_wmma_f32_16x16x4_f32(false, a0, false, b0,
                                                          (short)0, oacc0, false, false);
            oacc1 = __builtin_amdgcn_wmma_f32_16x16x4_f32(false, a1, false, b1,
                                                          (short)0, oacc1, false, false);
        }
    }

    // ---------------- write weighted values (16x64), coalesced 64B per lane-half ----------------
    {
        float* o = out_wv + ((size_t)bh * S_LEN + qb) * D_HEAD + wcol + m;
        #pragma unroll
        for (int i = 0; i < 8; ++i)
            o[(size_t)(i + 8 * h) * D_HEAD] = oacc0[i] + oacc1[i];
    }
}

extern "C" void kernel_launch(void* const* d_in, const int* in_sizes, int n_in,
                              void* d_out, int out_size, void* d_ws, size_t ws_size,
                              hipStream_t stream) {
    const float*         q    = (const float*)d_in[0];
    const float*         k    = (const float*)d_in[1];
    const float*         v    = (const float*)d_in[2];
    const unsigned char* mask = (const unsigned char*)d_in[3];  // jnp bool -> 1 byte

    const int B = 2, H = 16;
    float* wv    = (float*)d_out;                                   // [B,H,S,D] first
    float* probs = (float*)d_out + (size_t)B * H * S_LEN * D_HEAD;  // then [B,H,S,S]

    dim3 grid(B * H * (S_LEN / QT));   // 4096 workgroups, one 16-query tile each
    attn_fwd_f32wmma_kernel<<<grid, NTHREADS, 0, stream>>>(q, k, v, mask, wv, probs);
}